// SiameseGNN_37177236914659
// MI455X (gfx1250) — compile-verified
//
#include <hip/hip_runtime.h>
#include <hip/hip_bf16.h>
#include <math.h>

typedef __attribute__((ext_vector_type(2))) float v2f;
typedef __attribute__((ext_vector_type(8))) float v8f;

#define DIN   128
#define DH1   128
#define DH2   256
#define TOPK  512

__device__ __forceinline__ void atomAddF(float* p, float v) {
    __hip_atomic_fetch_add(p, v, __ATOMIC_RELAXED, __HIP_MEMORY_SCOPE_AGENT);
}

// ---------------- utility kernels ----------------

__global__ void zero_kernel(float* __restrict__ p, size_t n) {
    size_t i = (size_t)blockIdx.x * blockDim.x + threadIdx.x;
    size_t stride = (size_t)gridDim.x * blockDim.x;
    for (; i < n; i += stride) p[i] = 0.0f;
}

__global__ void count_kernel(const int* __restrict__ dst, int* __restrict__ cnt, int ne) {
    int i = blockIdx.x * blockDim.x + threadIdx.x;
    if (i < ne)
        __hip_atomic_fetch_add(&cnt[dst[i]], 1, __ATOMIC_RELAXED, __HIP_MEMORY_SCOPE_AGENT);
}

__global__ void invc_kernel(const int* __restrict__ cnt, float* __restrict__ invc, int n) {
    int i = blockIdx.x * blockDim.x + threadIdx.x;
    if (i < n) invc[i] = 1.0f / fmaxf((float)cnt[i], 1.0f);
}

// Wt[n][k] = W[k][n]; i runs over dout*din (Wt is dout x din, row-major)
__global__ void transpose_kernel(const float* __restrict__ W, float* __restrict__ Wt,
                                 int din, int dout) {
    int i = blockIdx.x * blockDim.x + threadIdx.x;
    if (i < din * dout) {
        int n = i / din, k = i - n * din;
        Wt[i] = W[k * dout + n];
    }
}

// one wave per edge: 32 lanes x float4 = 128 floats
__global__ void scatter_kernel(const float* __restrict__ x, const int* __restrict__ src,
                               const int* __restrict__ dst, float* __restrict__ agg, int ne) {
    int wid = blockIdx.x * (blockDim.x >> 5) + (threadIdx.x >> 5);
    if (wid >= ne) return;
    int lane = threadIdx.x & 31;
    int s = src[wid], t = dst[wid];
    const float4 v = ((const float4*)(x + (size_t)s * DIN))[lane];
    float* ap = agg + (size_t)t * DIN + lane * 4;
    atomAddF(ap + 0, v.x); atomAddF(ap + 1, v.y);
    atomAddF(ap + 2, v.z); atomAddF(ap + 3, v.w);
}

// agg[row][*] *= invc[row]   (DIN==128 -> row = i>>7)
__global__ void scale_rows_kernel(float* __restrict__ agg, const float* __restrict__ invc, size_t total) {
    size_t i = (size_t)blockIdx.x * blockDim.x + threadIdx.x;
    size_t stride = (size_t)gridDim.x * blockDim.x;
    for (; i < total; i += stride) agg[i] *= invc[i >> 7];
}

// ---------------- fused SAGE layer: y = relu(aggm @ Wl + b + x @ Wr) ----------------
// One wave computes a 16x32 tile (two 16x16 accumulators) with V_WMMA_F32_16X16X4_F32.
// Weights are pre-transposed (Wt[n][k]) so B fragments are contiguous b64 loads;
// A fragments are shared across both N-tiles.
template <int DOUT, bool DIFF>
__global__ __launch_bounds__(256) void sage_gemm_kernel(
        const float* __restrict__ aggs,   // N x din (pre-scaled by 1/cnt)
        const float* __restrict__ x,      // N x din
        const float* __restrict__ Wlt,    // DOUT x din (transposed)
        const float* __restrict__ Wrt,    // DOUT x din (transposed)
        const float* __restrict__ bias,   // DOUT
        float* __restrict__ y,            // N x DOUT   (if !DIFF)
        const float* __restrict__ other,  // out1       (if DIFF)
        float* __restrict__ simsq,        // N          (if DIFF)
        int nrows, int din)
{
    const int NG = DOUT / 32;            // column groups of 32
    int wave = blockIdx.x * (blockDim.x >> 5) + (threadIdx.x >> 5);
    int mt = wave / NG;
    int ng = wave % NG;
    if (mt * 16 >= nrows) return;        // wave-uniform: EXEC stays all-ones

    int lane = threadIdx.x & 31;
    int half = lane >> 4;                // 0: K=k,k+1   1: K=k+2,k+3
    int lm   = lane & 15;
    int arow  = mt * 16 + lm;            // A-fragment row for this lane
    int bcol0 = ng * 32 + lm;            // B-fragment col, tile 0
    int bcol1 = bcol0 + 16;              // B-fragment col, tile 1

    const float* __restrict__ aggp = aggs + (size_t)arow * din;
    const float* __restrict__ xp   = x    + (size_t)arow * din;
    const float* __restrict__ wl0  = Wlt + (size_t)bcol0 * din;
    const float* __restrict__ wl1  = Wlt + (size_t)bcol1 * din;
    const float* __restrict__ wr0  = Wrt + (size_t)bcol0 * din;
    const float* __restrict__ wr1  = Wrt + (size_t)bcol1 * din;

    v8f acc0 = {};
    v8f acc1 = {};
    #pragma unroll 2
    for (int k = 0; k < din; k += 4) {
        int ka = k + half * 2;
        v2f a1  = *(const v2f*)(aggp + ka);   // b64
        v2f a2  = *(const v2f*)(xp   + ka);   // b64
        v2f bl0 = *(const v2f*)(wl0  + ka);   // b64 (transposed weights)
        v2f bl1 = *(const v2f*)(wl1  + ka);
        v2f br0 = *(const v2f*)(wr0  + ka);
        v2f br1 = *(const v2f*)(wr1  + ka);
        acc0 = __builtin_amdgcn_wmma_f32_16x16x4_f32(false, a1, false, bl0, (short)0, acc0, false, false);
        acc1 = __builtin_amdgcn_wmma_f32_16x16x4_f32(false, a1, false, bl1, (short)0, acc1, false, false);
        acc0 = __builtin_amdgcn_wmma_f32_16x16x4_f32(false, a2, false, br0, (short)0, acc0, false, false);
        acc1 = __builtin_amdgcn_wmma_f32_16x16x4_f32(false, a2, false, br1, (short)0, acc1, false, false);
    }

    int ncol0 = ng * 32 + lm;            // C/D: N = lane (mod 16)
    int ncol1 = ncol0 + 16;
    float bn0 = bias[ncol0];
    float bn1 = bias[ncol1];
    #pragma unroll
    for (int r = 0; r < 8; ++r) {        // C/D: M = r + 8*half
        int m = mt * 16 + r + half * 8;
        float v0 = fmaxf(acc0[r] + bn0, 0.0f);
        float v1 = fmaxf(acc1[r] + bn1, 0.0f);
        if constexpr (DIFF) {
            float d0 = other[(size_t)m * DOUT + ncol0] - v0 + 1e-6f;
            float d1 = other[(size_t)m * DOUT + ncol1] - v1 + 1e-6f;
            atomAddF(&simsq[m], d0 * d0 + d1 * d1);
        } else {
            y[(size_t)m * DOUT + ncol0] = v0;
            y[(size_t)m * DOUT + ncol1] = v1;
        }
    }
}

// ---------------- top-512 (radix select + bitonic) + fused MLP head ----------------

__global__ __launch_bounds__(1024) void topk_mlp_kernel(
        const float* __restrict__ simsq, int n,
        const float* __restrict__ fc1W, const float* __restrict__ fc1b,
        const float* __restrict__ bn1g, const float* __restrict__ bn1b,
        const float* __restrict__ fc2W, const float* __restrict__ fc2b,
        const float* __restrict__ bn2g, const float* __restrict__ bn2b,
        const float* __restrict__ fc3W, const float* __restrict__ fc3b,
        float* __restrict__ out)
{
    __shared__ unsigned hist[256];
    __shared__ unsigned s_prefix, s_K, s_cnt;
    __shared__ float top[TOPK];
    __shared__ float h1s[DH1];
    __shared__ float h2s[DH2];
    __shared__ float red[256];
    int tid = threadIdx.x;

    if (tid == 0) { s_prefix = 0u; s_K = TOPK; }
    unsigned prefmask = 0u;
    __syncthreads();

    // radix-select the 512th-largest key (monotone float->uint mapping)
    for (int shift = 24; shift >= 0; shift -= 8) {
        if (tid < 256) hist[tid] = 0u;
        __syncthreads();
        unsigned pref = s_prefix;
        for (int i = tid; i < n; i += blockDim.x) {
            float f = sqrtf(simsq[i]);
            unsigned b = __float_as_uint(f);
            unsigned u = b ^ ((b >> 31) ? 0xFFFFFFFFu : 0x80000000u);
            if ((u & prefmask) == pref)
                atomicAdd(&hist[(u >> shift) & 255u], 1u);
        }
        __syncthreads();
        if (tid == 0) {
            unsigned K = s_K, cum = 0u; int digit = 0;
            for (int d = 255; d >= 0; --d) {
                if (cum + hist[d] >= K) { digit = d; break; }
                cum += hist[d];
            }
            s_K = K - cum;
            s_prefix = pref | ((unsigned)digit << shift);
        }
        __syncthreads();
        prefmask |= (0xFFu << shift);
    }
    unsigned Tu = s_prefix;              // key of the 512th largest

    // gather strictly-greater values, pad with threshold copies
    if (tid == 0) s_cnt = 0u;
    __syncthreads();
    for (int i = tid; i < n; i += blockDim.x) {
        float f = sqrtf(simsq[i]);
        unsigned b = __float_as_uint(f);
        unsigned u = b ^ ((b >> 31) ? 0xFFFFFFFFu : 0x80000000u);
        if (u > Tu) {
            unsigned pos = atomicAdd(&s_cnt, 1u);
            if (pos < TOPK) top[pos] = f;
        }
    }
    __syncthreads();
    unsigned G = s_cnt;                  // #strictly greater (= 512 - s_K)
    float tv = __uint_as_float((Tu >> 31) ? (Tu ^ 0x80000000u) : ~Tu);
    for (int i = (int)G + tid; i < TOPK; i += blockDim.x) top[i] = tv;
    __syncthreads();

    // bitonic sort ascending (read reversed later for descending top-k order)
    for (int k = 2; k <= TOPK; k <<= 1) {
        for (int j = k >> 1; j > 0; j >>= 1) {
            if (tid < TOPK) {
                int ixj = tid ^ j;
                if (ixj > tid) {
                    float a = top[tid], c = top[ixj];
                    bool up = ((tid & k) == 0);
                    if (up ? (a > c) : (a < c)) { top[tid] = c; top[ixj] = a; }
                }
            }
            __syncthreads();
        }
    }

    const float bnscale = rsqrtf(1.0f + 1e-5f);
    if (tid < DH1) {
        float s = fc1b[tid];
        for (int k2 = 0; k2 < TOPK; ++k2) s += top[TOPK - 1 - k2] * fc1W[k2 * DH1 + tid];
        s = s * (bn1g[tid] * bnscale) + bn1b[tid];
        h1s[tid] = fmaxf(s, 0.0f);
    }
    __syncthreads();
    if (tid < DH2) {
        float s = fc2b[tid];
        for (int k2 = 0; k2 < DH1; ++k2) s += h1s[k2] * fc2W[k2 * DH2 + tid];
        s = s * (bn2g[tid] * bnscale) + bn2b[tid];
        h2s[tid] = fmaxf(s, 0.0f);
    }
    __syncthreads();
    if (tid < 256) red[tid] = h2s[tid] * fc3W[tid];
    __syncthreads();
    for (int s2 = 128; s2 > 0; s2 >>= 1) {
        if (tid < s2) red[tid] += red[tid + s2];
        __syncthreads();
    }
    if (tid == 0) out[0] = 1.0f / (1.0f + expf(-(red[0] + fc3b[0])));
}

// ---------------- host orchestration ----------------

extern "C" void kernel_launch(void* const* d_in, const int* in_sizes, int n_in,
                              void* d_out, int out_size, void* d_ws, size_t ws_size,
                              hipStream_t stream) {
    const float* x1  = (const float*)d_in[0];
    const int*   e1  = (const int*)  d_in[1];
    const float* x2  = (const float*)d_in[2];
    const int*   e2  = (const int*)  d_in[3];
    const float* W1l = (const float*)d_in[4];
    const float* W1r = (const float*)d_in[5];
    const float* b1  = (const float*)d_in[6];
    const float* W2l = (const float*)d_in[7];
    const float* W2r = (const float*)d_in[8];
    const float* b2  = (const float*)d_in[9];
    const float* fc1W = (const float*)d_in[10];
    const float* fc1b = (const float*)d_in[11];
    const float* bn1g = (const float*)d_in[12];
    const float* bn1b = (const float*)d_in[13];
    const float* fc2W = (const float*)d_in[14];
    const float* fc2b = (const float*)d_in[15];
    const float* bn2g = (const float*)d_in[16];
    const float* bn2b = (const float*)d_in[17];
    const float* fc3W = (const float*)d_in[18];
    const float* fc3b = (const float*)d_in[19];

    const int N  = in_sizes[0] / DIN;    // 50000
    const int E  = in_sizes[1] / 2;      // 800000
    float* ws = (float*)d_ws;

    float* simsq = ws;                       // N
    int*   cnt   = (int*)(ws + N);           // N
    float* invc  = ws + 2 * (size_t)N;       // N
    float* agg   = ws + 3 * (size_t)N;       // N*128
    float* h     = agg + (size_t)N * DIN;    // N*128
    float* out1  = h   + (size_t)N * DIN;    // N*256
    float* w1lt  = out1 + (size_t)N * DH2;   // 128*128
    float* w1rt  = w1lt + DIN * DH1;         // 128*128
    float* w2lt  = w1rt + DIN * DH1;         // 256*128
    float* w2rt  = w2lt + DIN * DH2;         // 256*128

    const int ZB = 1024;
    const int mtiles = (N + 15) / 16;                              // 3125 exactly
    const int gemm_blocks_128 = (mtiles * (DH1 / 32) + 7) / 8;     // 8 waves/block
    const int gemm_blocks_256 = (mtiles * (DH2 / 32) + 7) / 8;
    const int scatter_blocks  = (E * 32 + 255) / 256;              // 1 wave/edge
    const size_t aggN = (size_t)N * DIN;

    // weight transposes (tiny, once per launch)
    transpose_kernel<<<(DIN * DH1 + 255) / 256, 256, 0, stream>>>(W1l, w1lt, DIN, DH1);
    transpose_kernel<<<(DIN * DH1 + 255) / 256, 256, 0, stream>>>(W1r, w1rt, DIN, DH1);
    transpose_kernel<<<(DIN * DH2 + 255) / 256, 256, 0, stream>>>(W2l, w2lt, DIN, DH2);
    transpose_kernel<<<(DIN * DH2 + 255) / 256, 256, 0, stream>>>(W2r, w2rt, DIN, DH2);

    zero_kernel<<<512, ZB, 0, stream>>>(simsq, (size_t)N);

    // -------- graph 1 --------
    zero_kernel<<<512, ZB, 0, stream>>>((float*)cnt, (size_t)N);
    count_kernel<<<(E + 255) / 256, 256, 0, stream>>>(e1 + E, cnt, E);
    invc_kernel<<<(N + 255) / 256, 256, 0, stream>>>(cnt, invc, N);

    zero_kernel<<<2048, ZB, 0, stream>>>(agg, aggN);
    scatter_kernel<<<scatter_blocks, 256, 0, stream>>>(x1, e1, e1 + E, agg, E);
    scale_rows_kernel<<<2048, ZB, 0, stream>>>(agg, invc, aggN);
    sage_gemm_kernel<DH1, false><<<gemm_blocks_128, 256, 0, stream>>>(
        agg, x1, w1lt, w1rt, b1, h, nullptr, nullptr, N, DIN);

    zero_kernel<<<2048, ZB, 0, stream>>>(agg, aggN);
    scatter_kernel<<<scatter_blocks, 256, 0, stream>>>(h, e1, e1 + E, agg, E);
    scale_rows_kernel<<<2048, ZB, 0, stream>>>(agg, invc, aggN);
    sage_gemm_kernel<DH2, false><<<gemm_blocks_256, 256, 0, stream>>>(
        agg, h, w2lt, w2rt, b2, out1, nullptr, nullptr, N, DH1);

    // -------- graph 2 (layer-2 fused with squared-diff accumulation) --------
    zero_kernel<<<512, ZB, 0, stream>>>((float*)cnt, (size_t)N);
    count_kernel<<<(E + 255) / 256, 256, 0, stream>>>(e2 + E, cnt, E);
    invc_kernel<<<(N + 255) / 256, 256, 0, stream>>>(cnt, invc, N);

    zero_kernel<<<2048, ZB, 0, stream>>>(agg, aggN);
    scatter_kernel<<<scatter_blocks, 256, 0, stream>>>(x2, e2, e2 + E, agg, E);
    scale_rows_kernel<<<2048, ZB, 0, stream>>>(agg, invc, aggN);
    sage_gemm_kernel<DH1, false><<<gemm_blocks_128, 256, 0, stream>>>(
        agg, x2, w1lt, w1rt, b1, h, nullptr, nullptr, N, DIN);

    zero_kernel<<<2048, ZB, 0, stream>>>(agg, aggN);
    scatter_kernel<<<scatter_blocks, 256, 0, stream>>>(h, e2, e2 + E, agg, E);
    scale_rows_kernel<<<2048, ZB, 0, stream>>>(agg, invc, aggN);
    sage_gemm_kernel<DH2, true><<<gemm_blocks_256, 256, 0, stream>>>(
        agg, h, w2lt, w2rt, b2, nullptr, out1, simsq, N, DH1);

    // -------- top-512 + MLP head --------
    topk_mlp_kernel<<<1, 1024, 0, stream>>>(simsq, N,
        fc1W, fc1b, bn1g, bn1b, fc2W, fc2b, bn2g, bn2b, fc3W, fc3b,
        (float*)d_out);
}